// NeuralAutomaton_58454504898764
// MI455X (gfx1250) — compile-verified
//
#include <hip/hip_runtime.h>

// ============================================================================
// NeuralAutomaton on MI455X (gfx1250):
//   s_final = start @ (PROD_t W_t) @ accept,  W_t = sum_k x[t,k] * P[k]
// Ordered product-reduction of 2M stochastic 10x10 matrices, padded to 16x16,
// computed entirely with V_WMMA_F32_16X16X4_F32 (fp32 preserves reference
// precision; compute demand is tiny vs matrix peak, so no need for low prec).
//
//   Phase A: per-wave 1024-step chunks. Per 16 steps:
//            build D = X16[16x10] @ Pflat[10x100]  (21 WMMAs, P in VGPRs),
//            scatter D rows into 16 LDS matrices, tree-reduce 16->1
//            (15 products x 3 WMMAs), fold into running product M.
//   Phase B: 16->1 product trees over global chunk matrices (2048->128->8).
//   Phase C: 8->1 + start/accept contraction -> 2 floats.
//
// Layout-role trick: in the in-place tree, slot s produced at level `step` is
// consumed as the A operand iff s % (4*step) == 0, else as B. Results are
// stored directly in their consumer's layout:
//   A-destined -> row-major  (A slice k-pairs contiguous -> ds_load_b64)
//   B-destined -> col-major  (B slice k-pairs contiguous -> ds_load_b64,
//                             and the D write becomes 4x ds_store_b64)
// so no cross-lane permutes and no VGPR repacking moves are ever needed.
// Rows are padded to stride 18 floats: conflict-free-ish banks, 8B alignment.
// ============================================================================

#define T_STEPS  2097152
#define NSYM     10
#define NST      10
#define CHUNK    1024
#define NCHUNK   (T_STEPS / CHUNK)   // 2048
#define GROUPS   (CHUNK / 16)        // 64
#define WAVES    2                   // waves per block (static LDS < 64KB)
#define STRIDE   18                  // padded LDS row stride in floats
#define MSLOT    (16 * STRIDE)       // 288 floats per 16x16 matrix slot

typedef __attribute__((ext_vector_type(2))) float v2f;
typedef __attribute__((ext_vector_type(4))) float v4f;
typedef __attribute__((ext_vector_type(8))) float v8f;

// C = A @ B on padded 16x16 matrices in LDS (offsets in floats).
// K runs 0..11 (rows/cols >= 10 are zero), so 3 WMMA K-slices suffice.
// A is stored row-major, B col-major: both operand slices load as v2f at
// base + m*STRIDE + 4*kb + 2*h  (lane m = L%16, h = L/16, vgpr j contiguous).
// D (vgpr r -> element (r+8h, m)) stores row-major (8x b32) or col-major
// (4x b64) according to the consumer's role.
__device__ __forceinline__ void mat_product(float* lds, int outOfs, int aOfs,
                                            int bOfs, int m, int h,
                                            bool storeCol) {
  const int lo = m * STRIDE + 2 * h;
  v2f a0 = *(const v2f*)&lds[aOfs + lo + 0];
  v2f a1 = *(const v2f*)&lds[aOfs + lo + 4];
  v2f a2 = *(const v2f*)&lds[aOfs + lo + 8];
  v2f b0 = *(const v2f*)&lds[bOfs + lo + 0];
  v2f b1 = *(const v2f*)&lds[bOfs + lo + 4];
  v2f b2 = *(const v2f*)&lds[bOfs + lo + 8];
  v8f acc = {0.f, 0.f, 0.f, 0.f, 0.f, 0.f, 0.f, 0.f};
  acc = __builtin_amdgcn_wmma_f32_16x16x4_f32(false, a0, false, b0, (short)0, acc, false, false);
  acc = __builtin_amdgcn_wmma_f32_16x16x4_f32(false, a1, false, b1, (short)0, acc, false, false);
  acc = __builtin_amdgcn_wmma_f32_16x16x4_f32(false, a2, false, b2, (short)0, acc, false, false);
  if (storeCol) {
#pragma unroll
    for (int r = 0; r < 8; r += 2) {
      v2f v = {acc[r], acc[r + 1]};
      *(v2f*)&lds[outOfs + m * STRIDE + 8 * h + r] = v;
    }
  } else {
#pragma unroll
    for (int r = 0; r < 8; ++r)
      lds[outOfs + (r + 8 * h) * STRIDE + m] = acc[r];
  }
}

// Ordered in-place product tree over N LDS slots; result in slot 0.
// combine(left, right) = left @ right preserves the time ordering.
// Intermediate store layouts follow the consumer-role rule; the final
// product's layout is chosen by the caller.
template <int N>
__device__ __forceinline__ void treeN(float* lds, int base, int m, int h,
                                      bool finalCol) {
#pragma unroll
  for (int step = 1; step < N; step *= 2) {
#pragma unroll
    for (int s = 0; s < N; s += 2 * step) {
      const bool col = (2 * step == N) ? finalCol : ((s % (4 * step)) != 0);
      mat_product(lds, base + s * MSLOT, base + s * MSLOT,
                  base + (s + step) * MSLOT, m, h, col);
    }
  }
}

// Emit a row-major LDS matrix to a global slot, optionally transposed so the
// global image is already in its downstream consumer's layout.
__device__ __forceinline__ void write_mat(const float* lds, int srcOfs,
                                          float* dst, bool colMajor, int lane) {
#pragma unroll
  for (int e = 0; e < 8; ++e) {
    const int idx = e * 32 + lane;
    const int i = idx >> 4, j = idx & 15;
    const float v = lds[srcOfs + i * STRIDE + j];
    dst[colMajor ? (j * STRIDE + i) : (i * STRIDE + j)] = v;
  }
}

// ---------------------------------------------------------------------------
// Kernel A: one wave per 1024-step chunk -> one 16x16 product matrix.
// ---------------------------------------------------------------------------
__global__ void __launch_bounds__(WAVES * 32)
chunk_kernel(const float* __restrict__ x, const float* __restrict__ P,
             float* __restrict__ out) {
  __shared__ float lds[WAVES * 17 * MSLOT];  // 16 W slots + 1 running-M slot
  const int tid  = threadIdx.x;
  const int wave = tid >> 5;
  const int lane = tid & 31;
  const int m = lane & 15;
  const int h = lane >> 4;
  const int wbase = wave * 17 * MSLOT;
  const int mbase = wbase + 16 * MSLOT;
  const int chunkId = blockIdx.x * WAVES + wave;

  // Zero this wave's LDS region once (pad regions must start zero), then
  // seed the running product M (kept row-major: it is always the A operand).
  for (int i = lane; i < 17 * MSLOT; i += 32) lds[wbase + i] = 0.0f;
  if (lane < 16) lds[mbase + lane * STRIDE + lane] = 1.0f;

  // Preload P in B-operand layout for the build GEMM:
  // pb[kb][tn] element j = Pflat[k = 4kb+2h+j][n = tn*16 + m], zero-padded.
  v2f pb[3][7];
#pragma unroll
  for (int kb = 0; kb < 3; ++kb) {
#pragma unroll
    for (int tn = 0; tn < 7; ++tn) {
      const int n  = tn * 16 + m;
      const int k0 = 4 * kb + 2 * h;
      v2f v = {0.0f, 0.0f};
      if (n < NST * NST) {
        if (k0 < NSYM)     v.x = P[k0 * (NST * NST) + n];
        if (k0 + 1 < NSYM) v.y = P[(k0 + 1) * (NST * NST) + n];
      }
      pb[kb][tn] = v;
    }
  }

  const long chunk_t0 = (long)chunkId * CHUNK;
  for (int g = 0; g < GROUPS; ++g) {
    const long t = chunk_t0 + g * 16 + m;  // this lane's timestep row
    if (g + 1 < GROUPS) __builtin_prefetch(&x[(t + 16) * NSYM], 0, 1);

    // A operand of build GEMM: xa[kb] element j = x[t][4kb+2h+j] (contiguous,
    // 8B aligned since 4kb+2h is even and rows are 10 floats).
    v2f xa[3];
#pragma unroll
    for (int kb = 0; kb < 3; ++kb) {
      const int s0 = 4 * kb + 2 * h;
      v2f v = {0.0f, 0.0f};
      if (s0 < NSYM) v = *(const v2f*)&x[t * NSYM + s0];
      xa[kb] = v;
    }

    // D = X16 @ Pflat; row `slot` of D is W_{t0+slot} flattened (n = 10i+j).
    // Scatter into LDS slot `slot`; even slots become A operands of the
    // level-0 tree products (row-major), odd slots become B (col-major).
#pragma unroll
    for (int tn = 0; tn < 7; ++tn) {
      v8f acc = {0.f, 0.f, 0.f, 0.f, 0.f, 0.f, 0.f, 0.f};
      acc = __builtin_amdgcn_wmma_f32_16x16x4_f32(false, xa[0], false, pb[0][tn], (short)0, acc, false, false);
      acc = __builtin_amdgcn_wmma_f32_16x16x4_f32(false, xa[1], false, pb[1][tn], (short)0, acc, false, false);
      acc = __builtin_amdgcn_wmma_f32_16x16x4_f32(false, xa[2], false, pb[2][tn], (short)0, acc, false, false);
      const int flatn = tn * 16 + m;
      if (flatn < NST * NST) {
        const int wi = flatn / NST, wj = flatn % NST;
        const int rowAddr = wi * STRIDE + wj;   // A-destined (even slot)
        const int colAddr = wj * STRIDE + wi;   // B-destined (odd slot)
#pragma unroll
        for (int r = 0; r < 8; ++r) {
          const int slot = r + 8 * h;           // D row = timestep in group
          lds[wbase + slot * MSLOT + ((r & 1) ? colAddr : rowAddr)] = acc[r];
        }
      }
    }

    treeN<16>(lds, wbase, m, h, /*finalCol=*/true);  // G -> slot 0, col-major
    mat_product(lds, mbase, mbase, wbase, m, h, /*storeCol=*/false);  // M=M@G
  }

  // Emit the chunk product; global layout chosen by the slot-parity role the
  // next reduction stage will use it in (even chunk -> A/row, odd -> B/col).
  write_mat(lds, mbase, out + (long)chunkId * MSLOT, (chunkId & 1) != 0, lane);
}

// ---------------------------------------------------------------------------
// Kernel B: each wave folds 16 consecutive global matrices (already stored in
// their role layouts) into one.
// ---------------------------------------------------------------------------
__global__ void __launch_bounds__(WAVES * 32)
reduce16_kernel(const float* __restrict__ in, float* __restrict__ out) {
  __shared__ float lds[WAVES * 16 * MSLOT];
  const int tid  = threadIdx.x;
  const int wave = tid >> 5;
  const int lane = tid & 31;
  const int m = lane & 15, h = lane >> 4;
  const int wbase = wave * 16 * MSLOT;
  const long grpId = (long)blockIdx.x * WAVES + wave;

  const float* src = in + grpId * 16 * MSLOT;
  for (int i = lane; i < (16 * MSLOT) / 4; i += 32)
    *(v4f*)&lds[wbase + i * 4] = *(const v4f*)&src[i * 4];

  // Final product stored in the layout this output's consumer needs.
  treeN<16>(lds, wbase, m, h, /*finalCol=*/(grpId & 1) != 0);

  float* dst = out + grpId * MSLOT;
#pragma unroll
  for (int i = 0; i < MSLOT / 32; ++i)
    dst[i * 32 + lane] = lds[wbase + i * 32 + lane];
}

// ---------------------------------------------------------------------------
// Kernel C: fold the last 8 matrices, contract with start/accept.
// ---------------------------------------------------------------------------
__global__ void __launch_bounds__(32)
final_kernel(const float* __restrict__ in, const float* __restrict__ start,
             const float* __restrict__ accept, float* __restrict__ outv) {
  __shared__ float lds[8 * MSLOT];
  const int lane = threadIdx.x;
  const int m = lane & 15, h = lane >> 4;

  for (int i = lane; i < (8 * MSLOT) / 4; i += 32)
    *(v4f*)&lds[i * 4] = *(const v4f*)&in[i * 4];

  treeN<8>(lds, 0, m, h, /*finalCol=*/false);  // row-major for contraction

  if (lane == 0) {  // out[c] = sum_ij start[i] * M[i][j] * accept[j][c]
    float o0 = 0.f, o1 = 0.f;
    for (int i = 0; i < NST; ++i) {
      const float si = start[i];
      for (int j = 0; j < NST; ++j) {
        const float w = si * lds[i * STRIDE + j];
        o0 += w * accept[j * 2 + 0];
        o1 += w * accept[j * 2 + 1];
      }
    }
    outv[0] = o0;
    outv[1] = o1;
  }
}

// ---------------------------------------------------------------------------
extern "C" void kernel_launch(void* const* d_in, const int* in_sizes, int n_in,
                              void* d_out, int out_size, void* d_ws, size_t ws_size,
                              hipStream_t stream) {
  const float* x      = (const float*)d_in[0];  // [T, 10]
  const float* P      = (const float*)d_in[1];  // [10, 10, 10]
  const float* start  = (const float*)d_in[2];  // [10]
  const float* accept = (const float*)d_in[3];  // [10, 2]

  // Scratch layout (floats): 2048 + 128 + 8 padded matrices (~2.5 MB total).
  float* w0 = (float*)d_ws;
  float* w1 = w0 + (size_t)NCHUNK * MSLOT;
  float* w2 = w1 + (size_t)(NCHUNK / 16) * MSLOT;

  hipLaunchKernelGGL(chunk_kernel, dim3(NCHUNK / WAVES), dim3(WAVES * 32), 0,
                     stream, x, P, w0);                       // 2M steps -> 2048
  hipLaunchKernelGGL(reduce16_kernel, dim3((NCHUNK / 16) / WAVES),
                     dim3(WAVES * 32), 0, stream, w0, w1);    // 2048 -> 128
  hipLaunchKernelGGL(reduce16_kernel, dim3((NCHUNK / 256) / WAVES),
                     dim3(WAVES * 32), 0, stream, w1, w2);    // 128 -> 8
  hipLaunchKernelGGL(final_kernel, dim3(1), dim3(32), 0, stream, w2, start,
                     accept, (float*)d_out);                  // 8 -> out[2]
}